// YOLOV3Loss_54125177864881
// MI455X (gfx1250) — compile-verified
//
#include <hip/hip_runtime.h>
#include <hip/hip_bf16.h>

typedef float v2f __attribute__((ext_vector_type(2)));
typedef float v8f __attribute__((ext_vector_type(8)));

#define DEV __device__ __forceinline__

constexpr int BB = 32;        // batch
constexpr int MM = 32;        // gt boxes per image
constexpr int AA = 3;         // anchors per scale
constexpr int CC = 80;        // classes
constexpr int CH = 5 + CC;    // 85 channels per anchor
constexpr float IOU_THR = 0.7f;
constexpr float L_OBJ = 1.0f, L_NOOBJ = 1.0f, L_COORD = 0.75f;
constexpr float EPSF = 1e-8f;

constexpr int G0 = 13, G1 = 26, G2 = 52;
constexpr int NB0 = AA * G0 * G0;   // 507
constexpr int NB1 = AA * G1 * G1;   // 2028
constexpr int NB2 = AA * G2 * G2;   // 8112
constexpr int OFF0 = 0;
constexpr int OFF1 = BB * NB0;          // 16224
constexpr int OFF2 = OFF1 + BB * NB1;   // 81120
constexpr int TOTAL = OFF2 + BB * NB2;  // 340704 assigned-flag bytes (16B aligned)

// block partial-sum bookkeeping (one float per block, no atomics)
constexpr int BLK = 256;
constexpr int GX0 = (NB0 + BLK - 1) / BLK;   // 2
constexpr int GX1 = (NB1 + BLK - 1) / BLK;   // 8
constexpr int GX2 = (NB2 + BLK - 1) / BLK;   // 32
constexpr int PB0 = 0;
constexpr int PB1 = PB0 + GX0 * BB;          // 64
constexpr int PB2 = PB1 + GX1 * BB;          // 320
constexpr int PBP = PB2 + GX2 * BB;          // 1344
constexpr int NPART = PBP + BB * MM;         // 2368

// v_rcp_f32 fast reciprocal (~1 ulp) instead of IEEE div sequence
DEV float frcp(float x) { return __builtin_amdgcn_rcpf(x); }
DEV float sigmoidf(float x) { return frcp(1.0f + __expf(-x)); }

// -------- Exact f32 wave(32) sum using V_WMMA_F32_16X16X4_F32 --------
// Stage 1: A = partials (VGPR0=p, VGPR1=0) -> D[m][*] = p_m + p_{m+16}.
// Stage 2: A = ones, B = pairs of D regs; 4 accumulating WMMAs fold all 16
//          rowsums into every accumulator element.
DEV float wave_sum_wmma(float p) {
  v2f a;    a[0] = p;    a[1] = 0.0f;
  v2f ones; ones[0] = 1.0f; ones[1] = 1.0f;
  v8f z = {0.f, 0.f, 0.f, 0.f, 0.f, 0.f, 0.f, 0.f};
  v8f d1 = __builtin_amdgcn_wmma_f32_16x16x4_f32(false, a, false, ones,
                                                 (short)0, z, false, false);
  v8f acc = z;
  v2f b;
  b[0] = d1[0]; b[1] = d1[1];
  acc = __builtin_amdgcn_wmma_f32_16x16x4_f32(false, ones, false, b,
                                              (short)0, acc, false, false);
  b[0] = d1[2]; b[1] = d1[3];
  acc = __builtin_amdgcn_wmma_f32_16x16x4_f32(false, ones, false, b,
                                              (short)0, acc, false, false);
  b[0] = d1[4]; b[1] = d1[5];
  acc = __builtin_amdgcn_wmma_f32_16x16x4_f32(false, ones, false, b,
                                              (short)0, acc, false, false);
  b[0] = d1[6]; b[1] = d1[7];
  acc = __builtin_amdgcn_wmma_f32_16x16x4_f32(false, ones, false, b,
                                              (short)0, acc, false, false);
  return acc[0];
}

// -------- Kernel 0: zero scratch flags --------
__global__ void k_init(unsigned char* __restrict__ assigned) {
  int i = blockIdx.x * blockDim.x + threadIdx.x;
  if (i < TOTAL) assigned[i] = 0;
}

// -------- Kernel 1: scatter "assigned" flags (one thread per GT) --------
__global__ void k_scatter(const float* __restrict__ gt_boxes,
                          const float* __restrict__ anchors,
                          unsigned char* __restrict__ assigned) {
  int i = blockIdx.x * blockDim.x + threadIdx.x;
  if (i >= BB * MM) return;
  int b = i / MM, m = i - b * MM;
  const float* gb = gt_boxes + (size_t)(b * MM + m) * 4;
  float cx = gb[0], cy = gb[1], gw = gb[2], gh = gb[3];
  int bp = 0; float best = -1.0f;
  for (int k = 0; k < 9; ++k) {          // first-max like jnp.argmax
    float aw = anchors[2 * k], ah = anchors[2 * k + 1];
    float inter = fminf(gw, aw) * fminf(gh, ah);
    float uni = gw * gh + aw * ah - inter;
    float r = inter * frcp(uni + EPSF);
    if (r > best) { best = r; bp = k; }
  }
  int s = bp / 3;
  int G    = (s == 0) ? G0   : ((s == 1) ? G1   : G2);
  int gofs = (s == 0) ? OFF0 : ((s == 1) ? OFF1 : OFF2);
  int NB = AA * G * G;
  int gx = (int)fminf(fmaxf(floorf(cx * (float)G), 0.0f), (float)(G - 1));
  int gy = (int)fminf(fmaxf(floorf(cy * (float)G), 0.0f), (float)(G - 1));
  int box_idx = (bp - 3 * s) * G * G + gy * G + gx;
  assigned[gofs + b * NB + box_idx] = 1;
}

// -------- Kernel 2: no-object loss over every predicted box of one scale ----
__global__ void __launch_bounds__(BLK)
k_noobj(const float* __restrict__ p,
        const float* __restrict__ gt_boxes,
        const float* __restrict__ anchors,
        const unsigned char* __restrict__ assigned,
        int G, float invG, int scale, int gofs,
        float* __restrict__ partials, int pbase) {
  __shared__ float sx1[MM], sy1[MM], sx2[MM], sy2[MM], sarea[MM];
  __shared__ float sw[BLK / 32];
  int tid = threadIdx.x;
  int b = blockIdx.y;
  if (tid < MM) {
    const float* gb = gt_boxes + (size_t)(b * MM + tid) * 4;
    float cx = gb[0], cy = gb[1], w = gb[2], h = gb[3];
    sx1[tid] = cx - 0.5f * w; sy1[tid] = cy - 0.5f * h;
    sx2[tid] = cx + 0.5f * w; sy2[tid] = cy + 0.5f * h;
    sarea[tid] = w * h;
  }
  __syncthreads();
  int GG = G * G;
  int NB = AA * GG;
  int j = blockIdx.x * blockDim.x + tid;
  float contrib = 0.0f;
  if (j < NB) {
    int a  = j / GG;
    int r  = j - a * GG;
    int gy = r / G;
    int gx = r - gy * G;
    const float* base = p + (size_t)((b * AA + a) * CH) * GG + r;
    float tx  = base[0];
    float ty  = base[(size_t)GG];
    float tw  = base[(size_t)2 * GG];
    float th  = base[(size_t)3 * GG];
    float tob = base[(size_t)4 * GG];
    float sxv = sigmoidf(tx), syv = sigmoidf(ty), sob = sigmoidf(tob);
    float aw = anchors[2 * (scale * AA + a)];
    float ah = anchors[2 * (scale * AA + a) + 1];
    float bw = aw * __expf(tw), bh = ah * __expf(th);
    float bcx = (sxv + (float)gx) * invG;
    float bcy = (syv + (float)gy) * invG;
    float px1 = bcx - 0.5f * bw, py1 = bcy - 0.5f * bh;
    float px2 = bcx + 0.5f * bw, py2 = bcy + 0.5f * bh;
    float parea = bw * bh;
    // iou > THR  <=>  inter > THR * denom  (denom > 0) : no division needed
    bool hi = false;
#pragma unroll 8
    for (int m = 0; m < MM; ++m) {
      float lx = fmaxf(px1, sx1[m]), ly = fmaxf(py1, sy1[m]);
      float rx = fminf(px2, sx2[m]), ry = fminf(py2, sy2[m]);
      float iw = fmaxf(rx - lx, 0.0f), ih = fmaxf(ry - ly, 0.0f);
      float inter = iw * ih;
      float denom = sarea[m] + parea - inter + EPSF;
      hi = hi || (inter > IOU_THR * denom);
    }
    bool positive = hi || (assigned[gofs + b * NB + j] != 0);
    if (!positive) contrib = sob * sob;        // positivity == 0 -> noobj term
  }
  float wsum = wave_sum_wmma(L_NOOBJ * contrib);
  if ((tid & 31) == 0) sw[tid >> 5] = wsum;
  __syncthreads();
  if (tid == 0) {
    float t = 0.0f;
#pragma unroll
    for (int w = 0; w < BLK / 32; ++w) t += sw[w];
    partials[pbase + blockIdx.y * gridDim.x + blockIdx.x] = t;
  }
}

// -------- Kernel 3: positive-box coord/obj/class losses (one block per GT) --
__global__ void __launch_bounds__(128)
k_pos(const float* __restrict__ p0, const float* __restrict__ p1,
      const float* __restrict__ p2,
      const float* __restrict__ gt_boxes, const int* __restrict__ gt_classes,
      const float* __restrict__ anchors,
      float* __restrict__ partials) {
  __shared__ float sw[4];
  int blk = blockIdx.x;                 // b*MM + m
  int b = blk / MM, m = blk - b * MM;
  int tid = threadIdx.x;

  const float* gb = gt_boxes + (size_t)(b * MM + m) * 4;
  float cx = gb[0], cy = gb[1], gw = gb[2], gh = gb[3];
  int bp = 0; float best = -1.0f;
  for (int k = 0; k < 9; ++k) {
    float aw = anchors[2 * k], ah = anchors[2 * k + 1];
    float inter = fminf(gw, aw) * fminf(gh, ah);
    float uni = gw * gh + aw * ah - inter;
    float r2 = inter * frcp(uni + EPSF);
    if (r2 > best) { best = r2; bp = k; }
  }
  int s = bp / 3;
  int G    = (s == 0) ? G0   : ((s == 1) ? G1   : G2);
  int gofs = (s == 0) ? OFF0 : ((s == 1) ? OFF1 : OFF2);
  const float* p = (s == 0) ? p0 : ((s == 1) ? p1 : p2);
  int GG = G * G, NB = AA * GG;
  float gxf = fminf(fmaxf(floorf(cx * (float)G), 0.0f), (float)(G - 1));
  float gyf = fminf(fmaxf(floorf(cy * (float)G), 0.0f), (float)(G - 1));
  int gx = (int)gxf, gy = (int)gyf;
  int arel = bp - 3 * s;
  int box_idx = arel * GG + gy * G + gx;
  int gidx = gofs + b * NB + box_idx;
  bool valid = (gidx != 0);             // reference quirk: row 0 is excluded

  const float* base = p + (size_t)((b * AA + arel) * CH) * GG + gy * G + gx;

  float term = 0.0f;
  if (valid) {
    if (tid < CC) {                               // class MSE vs one-hot
      float sc = sigmoidf(base[(size_t)(5 + tid) * GG]);
      float tgt = (gt_classes[b * MM + m] == tid) ? 1.0f : 0.0f;
      float d = sc - tgt;
      term = d * d;
    } else if (tid == CC) {                       // objectness
      float sob = sigmoidf(base[(size_t)4 * GG]);
      float d = sob - 1.0f;
      term = L_OBJ * d * d;
    } else if (tid >= CC + 1 && tid <= CC + 4) {  // coord (per-component)
      int k = tid - (CC + 1);
      float w = 2.0f - gw * gh;
      float pr, t;
      if (k == 0)      { pr = sigmoidf(base[0]);                 t = cx * (float)G - gxf; }
      else if (k == 1) { pr = sigmoidf(base[(size_t)GG]);        t = cy * (float)G - gyf; }
      else if (k == 2) { pr = base[(size_t)2 * GG];
                         t = __logf(fmaxf(gw, EPSF)) - __logf(anchors[2 * bp]); }
      else             { pr = base[(size_t)3 * GG];
                         t = __logf(fmaxf(gh, EPSF)) - __logf(anchors[2 * bp + 1]); }
      float d = pr - t;
      term = L_COORD * w * d * d;
    }
  }
  float wsum = wave_sum_wmma(term);
  if ((tid & 31) == 0) sw[tid >> 5] = wsum;
  __syncthreads();
  if (tid == 0) {
    float t = sw[0] + sw[1] + sw[2] + sw[3];
    partials[PBP + blk] = t;
  }
}

// -------- Kernel 4: deterministic final reduction (single block) --------
__global__ void __launch_bounds__(256)
k_reduce(const float* __restrict__ partials, float* __restrict__ out) {
  __shared__ float sw[8];
  int tid = threadIdx.x;
  float s = 0.0f;
  for (int i = tid; i < NPART; i += 256) s += partials[i];
  float wsum = wave_sum_wmma(s);
  if ((tid & 31) == 0) sw[tid >> 5] = wsum;
  __syncthreads();
  if (tid == 0) {
    float t = 0.0f;
#pragma unroll
    for (int w = 0; w < 8; ++w) t += sw[w];
    out[0] = t;
  }
}

extern "C" void kernel_launch(void* const* d_in, const int* in_sizes, int n_in,
                              void* d_out, int out_size, void* d_ws, size_t ws_size,
                              hipStream_t stream) {
  (void)in_sizes; (void)n_in; (void)out_size; (void)ws_size;
  const float* p0      = (const float*)d_in[0];
  const float* p1      = (const float*)d_in[1];
  const float* p2      = (const float*)d_in[2];
  const float* gt      = (const float*)d_in[3];
  const int*   gtc     = (const int*)d_in[4];
  const float* anchors = (const float*)d_in[5];
  float* out = (float*)d_out;
  unsigned char* assigned = (unsigned char*)d_ws;
  float* partials = (float*)((char*)d_ws + TOTAL);   // TOTAL is 16B-aligned

  k_init<<<(TOTAL + BLK - 1) / BLK, BLK, 0, stream>>>(assigned);
  k_scatter<<<(BB * MM + BLK - 1) / BLK, BLK, 0, stream>>>(gt, anchors, assigned);

  dim3 g0(GX0, BB);
  k_noobj<<<g0, BLK, 0, stream>>>(p0, gt, anchors, assigned,
                                  G0, 1.0f / G0, 0, OFF0, partials, PB0);
  dim3 g1(GX1, BB);
  k_noobj<<<g1, BLK, 0, stream>>>(p1, gt, anchors, assigned,
                                  G1, 1.0f / G1, 1, OFF1, partials, PB1);
  dim3 g2(GX2, BB);
  k_noobj<<<g2, BLK, 0, stream>>>(p2, gt, anchors, assigned,
                                  G2, 1.0f / G2, 2, OFF2, partials, PB2);

  k_pos<<<BB * MM, 128, 0, stream>>>(p0, p1, p2, gt, gtc, anchors, partials);

  k_reduce<<<1, 256, 0, stream>>>(partials, out);
}